// HeteroGraphSAGE_33303176413371
// MI455X (gfx1250) — compile-verified
//
#include <hip/hip_runtime.h>

// Problem sizes (fixed by the reference)
#define N_DRUG 20000
#define N_PROT 8000
#define D_DRUG 300
#define D_PROT 1280
#define HIDDEN 512
#define OUT_DIM 5
#define N_EDGE 500000
#define N_LBL 200000
#define D_CAT (D_DRUG + HIDDEN)   // 812

typedef __attribute__((ext_vector_type(2))) float v2f;
typedef __attribute__((ext_vector_type(8))) float v8f;

// ---- workspace layout (in floats) -----------------------------------------
#define WS_AGG   0                                    // [N_PROT, D_DRUG]
#define WS_CNT   (WS_AGG  + N_PROT * D_DRUG)          // [N_PROT]
#define WS_INV   (WS_CNT  + N_PROT)                   // [N_PROT]
#define WS_PEMB  (WS_INV  + N_PROT)                   // [N_PROT, HIDDEN]
#define WS_DOUT  (WS_PEMB + N_PROT * HIDDEN)          // [N_DRUG, OUT_DIM]
#define WS_POUT  (WS_DOUT + N_DRUG * OUT_DIM)         // [N_PROT, OUT_DIM]
#define WS_TOTAL (WS_POUT + N_PROT * OUT_DIM)         // 6,652,000 floats ~26.6MB

// ---------------------------------------------------------------------------
// 0) zero the accumulation region (agg + cnt) -- must happen every call
__global__ void zero_kernel(float* __restrict__ p, int n) {
    int i = blockIdx.x * blockDim.x + threadIdx.x;
    int stride = gridDim.x * blockDim.x;
    for (; i < n; i += stride) p[i] = 0.0f;
}

// ---------------------------------------------------------------------------
// 1) edge scatter: one wave per edge; lanes stride the 300-wide feature row.
//    x_drug (24MB) and agg (9.6MB) are L2-resident, so this is L2-bound.
__global__ __launch_bounds__(256)
void scatter_accum_kernel(const float* __restrict__ x_drug,
                          const int*   __restrict__ esrc,
                          const int*   __restrict__ edst,
                          float* __restrict__ agg,
                          float* __restrict__ cnt) {
    int wid  = (blockIdx.x * blockDim.x + threadIdx.x) >> 5;
    int lane = threadIdx.x & 31;
    if (wid >= N_EDGE) return;
    int s = esrc[wid];
    int t = edst[wid];
    const float* src = x_drug + (size_t)s * D_DRUG;
    float*       dst = agg    + (size_t)t * D_DRUG;
    for (int d = lane; d < D_DRUG; d += 32)
        atomicAdd(dst + d, src[d]);
    if (lane == 0) atomicAdd(cnt + t, 1.0f);
}

// ---------------------------------------------------------------------------
// 2) reciprocal of clipped counts (mean aggregation)
__global__ void inv_count_kernel(const float* __restrict__ cnt,
                                 float* __restrict__ inv) {
    int p = blockIdx.x * blockDim.x + threadIdx.x;
    if (p < N_PROT) inv[p] = 1.0f / fmaxf(cnt[p], 1.0f);
}

// ---------------------------------------------------------------------------
// 3) SAGE GEMM via V_WMMA_F32_16X16X4_F32:
//    pemb = (agg*inv) @ W_l^T + b_l + x_prot @ W_r^T        [N_PROT, HIDDEN]
//    One wave computes a 32x32 output tile = 2x2 fragments of 16x16,
//    reusing every A/B fragment load across two WMMAs.
//    A-frag layout (16x4 f32): lane = (k>=2 ? 16:0)+m, vgpr = k&1.
//    B-frag mirrored; C/D: vgpr i -> M = i + 8*(lane>>4), N = lane&15.
__global__ __launch_bounds__(256)
void sage_gemm_wmma_kernel(const float* __restrict__ agg,
                           const float* __restrict__ inv_cnt,
                           const float* __restrict__ x_prot,
                           const float* __restrict__ W_l,
                           const float* __restrict__ b_l,
                           const float* __restrict__ W_r,
                           float* __restrict__ pemb) {
    const int ROW_TILES = N_PROT / 32;   // 250
    const int COL_TILES = HIDDEN / 32;   // 16
    int wid = (blockIdx.x * blockDim.x + threadIdx.x) >> 5;
    if (wid >= ROW_TILES * COL_TILES) return;
    int rt = wid / COL_TILES;
    int ct = wid % COL_TILES;
    int r0 = rt * 32, c0 = ct * 32;

    int lane = threadIdx.x & 31;
    int mn   = lane & 15;        // M index (A/C) or N index (B/C)
    int g    = lane >> 4;        // lane group
    int koff = 2 * g;            // K sub-offset within a 4-wide K step

    v8f c00 = {}, c01 = {}, c10 = {}, c11 = {};

    // ---- phase 1: K over D_DRUG with A = mean(agg), B = W_l ----
    {
        const float* a0p = agg + (size_t)(r0 + mn)      * D_DRUG;
        const float* a1p = agg + (size_t)(r0 + 16 + mn) * D_DRUG;
        const float* b0p = W_l + (size_t)(c0 + mn)      * D_DRUG;
        const float* b1p = W_l + (size_t)(c0 + 16 + mn) * D_DRUG;
        float inv0 = inv_cnt[r0 + mn];
        float inv1 = inv_cnt[r0 + 16 + mn];
        for (int kk = 0; kk < D_DRUG; kk += 4) {
            int k = kk + koff;
            v2f a0 = { a0p[k] * inv0, a0p[k + 1] * inv0 };
            v2f a1 = { a1p[k] * inv1, a1p[k + 1] * inv1 };
            v2f b0 = { b0p[k],        b0p[k + 1] };
            v2f b1 = { b1p[k],        b1p[k + 1] };
            c00 = __builtin_amdgcn_wmma_f32_16x16x4_f32(false, a0, false, b0, (short)0, c00, false, false);
            c01 = __builtin_amdgcn_wmma_f32_16x16x4_f32(false, a0, false, b1, (short)0, c01, false, false);
            c10 = __builtin_amdgcn_wmma_f32_16x16x4_f32(false, a1, false, b0, (short)0, c10, false, false);
            c11 = __builtin_amdgcn_wmma_f32_16x16x4_f32(false, a1, false, b1, (short)0, c11, false, false);
        }
    }

    // ---- phase 2: K over D_PROT with A = x_prot, B = W_r ----
    {
        const float* a0p = x_prot + (size_t)(r0 + mn)      * D_PROT;
        const float* a1p = x_prot + (size_t)(r0 + 16 + mn) * D_PROT;
        const float* b0p = W_r    + (size_t)(c0 + mn)      * D_PROT;
        const float* b1p = W_r    + (size_t)(c0 + 16 + mn) * D_PROT;
        for (int kk = 0; kk < D_PROT; kk += 4) {
            int k = kk + koff;
            v2f a0 = { a0p[k], a0p[k + 1] };
            v2f a1 = { a1p[k], a1p[k + 1] };
            v2f b0 = { b0p[k], b0p[k + 1] };
            v2f b1 = { b1p[k], b1p[k + 1] };
            c00 = __builtin_amdgcn_wmma_f32_16x16x4_f32(false, a0, false, b0, (short)0, c00, false, false);
            c01 = __builtin_amdgcn_wmma_f32_16x16x4_f32(false, a0, false, b1, (short)0, c01, false, false);
            c10 = __builtin_amdgcn_wmma_f32_16x16x4_f32(false, a1, false, b0, (short)0, c10, false, false);
            c11 = __builtin_amdgcn_wmma_f32_16x16x4_f32(false, a1, false, b1, (short)0, c11, false, false);
        }
    }

    // ---- epilogue: add bias, store ----
    float bias0 = b_l[c0 + mn];
    float bias1 = b_l[c0 + 16 + mn];
    int rA = r0 + 8 * g;        // row base for c00/c01
    int rB = r0 + 16 + 8 * g;   // row base for c10/c11
#pragma unroll
    for (int i = 0; i < 8; ++i) {
        pemb[(size_t)(rA + i) * HIDDEN + c0 + mn]      = c00[i] + bias0;
        pemb[(size_t)(rA + i) * HIDDEN + c0 + 16 + mn] = c01[i] + bias1;
        pemb[(size_t)(rB + i) * HIDDEN + c0 + mn]      = c10[i] + bias0;
        pemb[(size_t)(rB + i) * HIDDEN + c0 + 16 + mn] = c11[i] + bias1;
    }
}

// ---------------------------------------------------------------------------
// 4) drug head: dout[d,o] = sum_k x_drug[d,k] * W_lin[o, k]   (W_lin tiny/L1)
__global__ void drug_head_kernel(const float* __restrict__ x_drug,
                                 const float* __restrict__ W_lin,
                                 float* __restrict__ dout) {
    int d = blockIdx.x * blockDim.x + threadIdx.x;
    if (d >= N_DRUG) return;
    const float* x = x_drug + (size_t)d * D_DRUG;
    float acc[OUT_DIM] = {0.f, 0.f, 0.f, 0.f, 0.f};
    for (int k = 0; k < D_DRUG; ++k) {
        float xv = x[k];
#pragma unroll
        for (int o = 0; o < OUT_DIM; ++o)
            acc[o] += xv * W_lin[o * D_CAT + k];
    }
#pragma unroll
    for (int o = 0; o < OUT_DIM; ++o)
        dout[(size_t)d * OUT_DIM + o] = acc[o];
}

// 5) protein head: pout[p,o] = sum_h pemb[p,h] * W_lin[o, 300+h]
__global__ void prot_head_kernel(const float* __restrict__ pemb,
                                 const float* __restrict__ W_lin,
                                 float* __restrict__ pout) {
    int p = blockIdx.x * blockDim.x + threadIdx.x;
    if (p >= N_PROT) return;
    const float* e = pemb + (size_t)p * HIDDEN;
    float acc[OUT_DIM] = {0.f, 0.f, 0.f, 0.f, 0.f};
    for (int h = 0; h < HIDDEN; ++h) {
        float ev = e[h];
#pragma unroll
        for (int o = 0; o < OUT_DIM; ++o)
            acc[o] += ev * W_lin[o * D_CAT + D_DRUG + h];
    }
#pragma unroll
    for (int o = 0; o < OUT_DIM; ++o)
        pout[(size_t)p * OUT_DIM + o] = acc[o];
}

// ---------------------------------------------------------------------------
// 6) labeled-edge output: out[l] = dout[src[l]] + pout[dst[l]] + b_lin
//    (exact linear decomposition of concat+Linear; head tables are L2-resident)
__global__ void edge_out_kernel(const int* __restrict__ lsrc,
                                const int* __restrict__ ldst,
                                const float* __restrict__ dout,
                                const float* __restrict__ pout,
                                const float* __restrict__ b_lin,
                                float* __restrict__ out) {
    int l = blockIdx.x * blockDim.x + threadIdx.x;
    if (l >= N_LBL) return;
    int s = lsrc[l];
    int t = ldst[l];
    const float* ds = dout + (size_t)s * OUT_DIM;
    const float* ps = pout + (size_t)t * OUT_DIM;
    float* op = out + (size_t)l * OUT_DIM;
#pragma unroll
    for (int o = 0; o < OUT_DIM; ++o)
        op[o] = ds[o] + ps[o] + b_lin[o];
}

// ---------------------------------------------------------------------------
extern "C" void kernel_launch(void* const* d_in, const int* in_sizes, int n_in,
                              void* d_out, int out_size, void* d_ws, size_t ws_size,
                              hipStream_t stream) {
    const float* x_drug = (const float*)d_in[0];
    const float* x_prot = (const float*)d_in[1];
    const int*   esrc   = (const int*)  d_in[2];
    const int*   edst   = (const int*)  d_in[3];
    const int*   lsrc   = (const int*)  d_in[4];
    const int*   ldst   = (const int*)  d_in[5];
    const float* W_l    = (const float*)d_in[6];
    const float* b_l    = (const float*)d_in[7];
    const float* W_r    = (const float*)d_in[8];
    const float* W_lin  = (const float*)d_in[9];
    const float* b_lin  = (const float*)d_in[10];
    float* out = (float*)d_out;

    float* ws   = (float*)d_ws;
    float* agg  = ws + WS_AGG;
    float* cnt  = ws + WS_CNT;
    float* inv  = ws + WS_INV;
    float* pemb = ws + WS_PEMB;
    float* dout = ws + WS_DOUT;
    float* pout = ws + WS_POUT;

    // 0) zero accumulators (agg + cnt contiguous)
    {
        int n = N_PROT * D_DRUG + N_PROT;
        int blocks = (n + 255) / 256;
        if (blocks > 4096) blocks = 4096;
        zero_kernel<<<blocks, 256, 0, stream>>>(agg, n);
    }
    // 1) scatter-add over edges (1 wave / edge)
    {
        long long threads = (long long)N_EDGE * 32;
        int blocks = (int)((threads + 255) / 256);
        scatter_accum_kernel<<<blocks, 256, 0, stream>>>(x_drug, esrc, edst, agg, cnt);
    }
    // 2) reciprocal counts
    inv_count_kernel<<<(N_PROT + 255) / 256, 256, 0, stream>>>(cnt, inv);
    // 3) WMMA GEMM -> prot_emb_all
    {
        int waves  = (N_PROT / 32) * (HIDDEN / 32);  // 4000
        int blocks = (waves * 32 + 255) / 256;       // 500
        sage_gemm_wmma_kernel<<<blocks, 256, 0, stream>>>(agg, inv, x_prot,
                                                          W_l, b_l, W_r, pemb);
    }
    // 4/5) head projections
    drug_head_kernel<<<(N_DRUG + 255) / 256, 256, 0, stream>>>(x_drug, W_lin, dout);
    prot_head_kernel<<<(N_PROT + 255) / 256, 256, 0, stream>>>(pemb, W_lin, pout);
    // 6) gather-add final output
    edge_out_kernel<<<(N_LBL + 255) / 256, 256, 0, stream>>>(lsrc, ldst, dout, pout,
                                                             b_lin, out);
    (void)in_sizes; (void)n_in; (void)out_size; (void)ws_size;
}